// StructNMaxPool2D_e_3728031613248
// MI455X (gfx1250) — compile-verified
//
#include <hip/hip_runtime.h>
#include <stdint.h>

// Problem geometry (fixed by the reference):
//   e, ce: [8, 5, 4, 512, 512] f32  ->  160 planes of 512x512
//   out:   e_out [160, 256, 256] then ce_out [160, 256, 256], concatenated.
#define H        512
#define W        512
#define HOUT     256
#define WOUT     256
#define PLANES   160          // B * C * D = 8 * 5 * 4
#define ROWS_BLK 8            // output rows per block
#define IN_ROWS  16           // input rows per block (2 * ROWS_BLK)
#define THREADS  256          // 8 wave32s
#define DEVALUE  0.25f        // 1 / (stride*stride)

__global__ __launch_bounds__(THREADS) void pool_gather_kernel(
    const float* __restrict__ e, const float* __restrict__ ce,
    float* __restrict__ out_e, float* __restrict__ out_ce)
{
    // 16 rows x 512 cols x 4 B x 2 arrays = 64 KB LDS tile.
    __shared__ __align__(16) float lds_e [IN_ROWS][W];
    __shared__ __align__(16) float lds_ce[IN_ROWS][W];

    const int t     = threadIdx.x;
    const int strip = blockIdx.x;           // 0..31  (output-row strip)
    const int plane = blockIdx.y;           // 0..159
    const int r0    = strip * IN_ROWS;      // first input row of the tile
    const int ro0   = strip * ROWS_BLK;     // first output row of the tile

    // Uniform 64-bit global bases (SGPRs) for the saddr form of the async op.
    const uint64_t gbase_e  = (uint64_t)(uintptr_t)(e  + ((size_t)plane * H + r0) * W);
    const uint64_t gbase_ce = (uint64_t)(uintptr_t)(ce + ((size_t)plane * H + r0) * W);
    // Low 32 bits of a generic pointer to LDS == LDS byte offset.
    const uint32_t lds_e_base  = (uint32_t)(uintptr_t)(&lds_e [0][0]);
    const uint32_t lds_ce_base = (uint32_t)(uintptr_t)(&lds_ce[0][0]);

    // --- Stage the 2 x 32 KB input tile into LDS with CDNA5 async copies. ---
    // The tile is a contiguous 32 KB span per array (16 full rows), so each
    // thread issues 8 dwordx4 async copies per array; consecutive lanes cover
    // consecutive 16 B -> perfectly coalesced 512 B bursts per wave.
#pragma unroll
    for (int k = 0; k < 8; ++k) {
        const uint32_t byte_off = (uint32_t)(t + k * THREADS) * 16u;
        const uint32_t ldse  = lds_e_base  + byte_off;
        const uint32_t ldsce = lds_ce_base + byte_off;
        asm volatile("global_load_async_to_lds_b128 %0, %1, %2"
                     :: "v"(ldse),  "v"(byte_off), "s"(gbase_e)  : "memory");
        asm volatile("global_load_async_to_lds_b128 %0, %1, %2"
                     :: "v"(ldsce), "v"(byte_off), "s"(gbase_ce) : "memory");
    }
    // Wait for this wave's async transfers, then sync all 8 waves of the block.
    asm volatile("s_wait_asynccnt 0" ::: "memory");
    __syncthreads();

    // --- Compute: each thread produces 8 consecutive output pixels. ---
    const int orow = t >> 5;                // 0..7 output row within tile
    const int oc8  = (t & 31) * 8;          // output col base, multiple of 8
    const int rr   = orow * 2;              // input row pair in LDS

    float er0[16], er1[16], cr0[16], cr1[16];
    const float4* pe0 = (const float4*)&lds_e [rr    ][oc8 * 2];
    const float4* pe1 = (const float4*)&lds_e [rr + 1][oc8 * 2];
    const float4* pc0 = (const float4*)&lds_ce[rr    ][oc8 * 2];
    const float4* pc1 = (const float4*)&lds_ce[rr + 1][oc8 * 2];
#pragma unroll
    for (int q = 0; q < 4; ++q) {           // 16 x ds_load_b128 per thread
        ((float4*)er0)[q] = pe0[q];
        ((float4*)er1)[q] = pe1[q];
        ((float4*)cr0)[q] = pc0[q];
        ((float4*)cr1)[q] = pc1[q];
    }

    float oe[8], oc[8];
#pragma unroll
    for (int i = 0; i < 8; ++i) {
        const float e00 = er0[2*i], e01 = er0[2*i+1];
        const float e10 = er1[2*i], e11 = er1[2*i+1];
        const float c00 = cr0[2*i], c01 = cr0[2*i+1];
        const float c10 = cr1[2*i], c11 = cr1[2*i+1];
        // Pool score, same as reference: ce * (1 - e)
        const float v00 = c00 * (1.0f - e00);
        const float v01 = c01 * (1.0f - e01);
        const float v10 = c10 * (1.0f - e10);
        const float v11 = c11 * (1.0f - e11);
        // First-occurrence argmax in order (r0,c0),(r0,c1),(r1,c0),(r1,c1):
        // strict '>' reproduces jnp.argmax tie-breaking.
        float best = v00, se = e00, sc = c00;
        if (v01 > best) { best = v01; se = e01; sc = c01; }
        if (v10 > best) { best = v10; se = e10; sc = c10; }
        if (v11 > best) { best = v11; se = e11; sc = c11; }
        oe[i] = se;
        oc[i] = sc * DEVALUE;
    }

    // --- B128 stores of both outputs. ---
    float* poe = out_e  + ((size_t)plane * HOUT + (ro0 + orow)) * WOUT + oc8;
    float* poc = out_ce + ((size_t)plane * HOUT + (ro0 + orow)) * WOUT + oc8;
    ((float4*)poe)[0] = ((const float4*)oe)[0];
    ((float4*)poe)[1] = ((const float4*)oe)[1];
    ((float4*)poc)[0] = ((const float4*)oc)[0];
    ((float4*)poc)[1] = ((const float4*)oc)[1];
}

extern "C" void kernel_launch(void* const* d_in, const int* in_sizes, int n_in,
                              void* d_out, int out_size, void* d_ws, size_t ws_size,
                              hipStream_t stream) {
    // Inputs in setup_inputs() order: d (unused), cd (unused), e, ce.
    const float* e  = (const float*)d_in[2];
    const float* ce = (const float*)d_in[3];
    float* out_e  = (float*)d_out;
    float* out_ce = (float*)d_out + (size_t)PLANES * HOUT * WOUT;

    dim3 grid(HOUT / ROWS_BLK, PLANES, 1);   // 32 x 160 = 5120 blocks
    dim3 block(THREADS, 1, 1);               // 8 wave32s
    pool_gather_kernel<<<grid, block, 0, stream>>>(e, ce, out_e, out_ce);
}